// LCC_55250459296434
// MI455X (gfx1250) — compile-verified
//
#include <hip/hip_runtime.h>

typedef __attribute__((ext_vector_type(2))) float v2f;
typedef __attribute__((ext_vector_type(8))) float v8f;

#define NVOX 192
#define TS   8            // output tile edge
#define DT   12           // residual tile edge (TS + 2*S)
#define IT   16           // input tile edge (TS + 4*S)
#define NB   (NVOX / TS)  // 24 blocks per axis
#define NBLK (NB * NB * NB)

#if __has_builtin(__builtin_amdgcn_global_load_async_to_lds_b32) && \
    __has_builtin(__builtin_amdgcn_s_wait_asynccnt)
#define HAVE_ASYNC_LDS 1
#else
#define HAVE_ASYNC_LDS 0
#endif

__device__ __forceinline__ int iclamp(int v, int lo, int hi) {
    return v < lo ? lo : (v > hi ? hi : v);
}

__global__ void zero_out_kernel(float* out) { out[0] = 0.0f; }

__global__ __launch_bounds__(512) void lcc_kernel(
    const float* __restrict__ imF, const float* __restrict__ imM,
    const float* __restrict__ mask, float* __restrict__ partials,
    float* __restrict__ out, int use_ws)
{
    __shared__ float sF[IT * IT * IT];
    __shared__ float sM[IT * IT * IT];
    __shared__ float dF[DT * DT * DT];
    __shared__ float dM[DT * DT * DT];
    __shared__ float red[16];

    const int tid = threadIdx.x;
    const int bw = blockIdx.x * TS, bh = blockIdx.y * TS, bd = blockIdx.z * TS;
    const int iw0 = bw - 4, ih0 = bh - 4, id0 = bd - 4;

    // Stage 0: load 16^3 edge-clamped input halos for both volumes into LDS.
    // gfx1250 async global->LDS (ASYNCcnt): data lands directly in LDS
    // without staging through VGPRs; fallback keeps plain loads.
    for (int r = tid; r < IT * IT * IT; r += 512) {  // exactly 8 uniform iters
        int w = r & 15, h = (r >> 4) & 15, d = r >> 8;
        int gw = iclamp(iw0 + w, 0, NVOX - 1);
        int gh = iclamp(ih0 + h, 0, NVOX - 1);
        int gd = iclamp(id0 + d, 0, NVOX - 1);
        long idx = ((long)gd * NVOX + gh) * NVOX + gw;
#if HAVE_ASYNC_LDS
        __builtin_amdgcn_global_load_async_to_lds_b32(
            (__attribute__((address_space(1))) int*)(imF + idx),
            (__attribute__((address_space(3))) int*)(&sF[r]),
            /*offset=*/0, /*cpol=*/0);
        __builtin_amdgcn_global_load_async_to_lds_b32(
            (__attribute__((address_space(1))) int*)(imM + idx),
            (__attribute__((address_space(3))) int*)(&sM[r]),
            /*offset=*/0, /*cpol=*/0);
#else
        sF[r] = imF[idx];
        sM[r] = imM[idx];
#endif
    }
#if HAVE_ASYNC_LDS
    __builtin_amdgcn_s_wait_asynccnt(0);
#endif
    __syncthreads();

    // Stage 1: residuals dcore = im - boxmean on the 12^3 halo region
    // (clamp the point coordinate first -> exact edge-replicate semantics).
    const float inv = 1.0f / 125.0f;
    for (int p = tid; p < DT * DT * DT; p += 512) {
        int pw = p % 12, ph = (p / 12) % 12, pd = p / 144;
        int lw = iclamp(bw - 2 + pw, 0, NVOX - 1) - iw0;
        int lh = iclamp(bh - 2 + ph, 0, NVOX - 1) - ih0;
        int ld = iclamp(bd - 2 + pd, 0, NVOX - 1) - id0;
        float aF = 0.0f, aM = 0.0f;
        for (int i = 0; i < 5; i++)
            for (int j = 0; j < 5; j++) {
                int b0 = ((ld - 2 + i) * IT + (lh - 2 + j)) * IT + (lw - 2);
#pragma unroll
                for (int k = 0; k < 5; k++) { aF += sF[b0 + k]; aM += sM[b0 + k]; }
            }
        int c0 = (ld * IT + lh) * IT + lw;
        dF[p] = sF[c0] - aF * inv;
        dM[p] = sM[c0] - aM * inv;
    }
    __syncthreads();

    // Stage 2: one output voxel per thread; 5^3 sums of products from LDS.
    int ow = tid & 7, oh = (tid >> 3) & 7, od = tid >> 6;
    float cr = 0.0f, vf = 0.0f, vm = 0.0f;
    for (int i = 0; i < 5; i++)
        for (int j = 0; j < 5; j++) {
            int b0 = ((od + i) * DT + (oh + j)) * DT + ow;
#pragma unroll
            for (int k = 0; k < 5; k++) {
                float fa = dF[b0 + k], fb = dM[b0 + k];
                cr += fa * fb;
                vf += fa * fa;
                vm += fb * fb;
            }
        }
    long g = ((long)(bd + od) * NVOX + (bh + oh)) * NVOX + (bw + ow);
    float lcc = cr * cr / (vf * vm + 1e-5f);
    float partial = -lcc * mask[g];

    // Stage 3: wave(32) reduction via V_WMMA_F32_16X16X4_F32.
    // A(16x4): lanes 0-15 supply K=0 (VGPR0), lanes 16-31 supply K=2; K=1,3 zero.
    // B(4x16) = ones  =>  D[m,n] = p[m] + p[m+16] for all n.
    // Lane n (<16) holds D rows 0..7, lane n+16 holds rows 8..15.
    v2f av; av.x = partial; av.y = 0.0f;
    v2f bv; bv.x = 1.0f;    bv.y = 1.0f;
    v8f cv = {};
    v8f dres = __builtin_amdgcn_wmma_f32_16x16x4_f32(
        /*neg_a=*/false, av, /*neg_b=*/false, bv,
        /*c_mod=*/(short)0, cv, /*reuse_a=*/false, /*reuse_b=*/false);
    float s = dres[0] + dres[1] + dres[2] + dres[3] +
              dres[4] + dres[5] + dres[6] + dres[7];
    s += __shfl_xor(s, 16, 32);   // combine halves -> full wave sum in all lanes

    int lane = tid & 31, wid = tid >> 5;
    if (lane == 0) red[wid] = s;
    __syncthreads();
    if (tid == 0) {
        float bs = 0.0f;
        for (int i = 0; i < 16; i++) bs += red[i];
        if (use_ws)
            partials[blockIdx.x + NB * (blockIdx.y + NB * blockIdx.z)] = bs;
        else
            atomicAdd(out, bs);   // fallback only if workspace is too small
    }
}

__global__ void finalize_kernel(const float* __restrict__ partials, float* out) {
    __shared__ float sh[256];
    int t = threadIdx.x;
    float acc = 0.0f;
    for (int i = t; i < NBLK; i += 256) acc += partials[i];
    sh[t] = acc;
    __syncthreads();
    for (int s = 128; s > 0; s >>= 1) {
        if (t < s) sh[t] += sh[t + s];
        __syncthreads();
    }
    if (t == 0) out[0] = sh[0];
}

extern "C" void kernel_launch(void* const* d_in, const int* in_sizes, int n_in,
                              void* d_out, int out_size, void* d_ws, size_t ws_size,
                              hipStream_t stream) {
    const float* imF  = (const float*)d_in[0];
    const float* imM  = (const float*)d_in[1];
    const float* mask = (const float*)d_in[2];
    float* out = (float*)d_out;
    float* partials = (float*)d_ws;

    int use_ws = (ws_size >= (size_t)NBLK * sizeof(float)) ? 1 : 0;

    zero_out_kernel<<<1, 1, 0, stream>>>(out);
    lcc_kernel<<<dim3(NB, NB, NB), 512, 0, stream>>>(imF, imM, mask,
                                                     partials, out, use_ws);
    if (use_ws)
        finalize_kernel<<<1, 256, 0, stream>>>(partials, out);
}